// scene_graph_with_monet_v2_62878321213570
// MI455X (gfx1250) — compile-verified
//
#include <hip/hip_runtime.h>
#include <hip/hip_bf16.h>
#include <cstddef>

// ---------------------------------------------------------------------------
// Problem constants (from the reference)
// ---------------------------------------------------------------------------
#define B_    2
#define S_    11
#define CIN_  512     // C
#define CREL_ 768     // C_REL
#define D_    256
#define MF_   32
#define H_    16
#define W_    24
#define HW_   384     // H*W
#define HM_   64
#define WM_   64
#define COBJ_ 1056    // C_OBJ_IN
#define CRIN_ 1344    // C_REL_IN
#define FCK_  98304   // D*H*W

typedef __attribute__((ext_vector_type(2))) float v2f;
typedef __attribute__((ext_vector_type(8))) float v8f;

// ---------------------------------------------------------------------------
// Wave-level fp32 WMMA GEMM block: D = A(64xK) * B(Kx16), K % 4 == 0.
// MT=4 M-tiles per wave: the (expensive, fused) B fragment is materialized
// once per K-step and reused by 4 WMMAs against 4 contiguous weight
// fragments -> ~1.25 loads per v_wmma instead of ~3.
// fp32 WMMA (V_WMMA_F32_16X16X4_F32) keeps full precision for this fp32 ref.
// A layout: lane l16 holds row M=l16; VGPR pair = K = {2*half, 2*half+1}
// B layout: lane l16 holds col N=l16; same K pair
// D layout: VGPR r <-> M = r + 8*half, N = l16
// ---------------------------------------------------------------------------
template <int MT, typename FA, typename FB, typename FS>
__device__ __forceinline__ void wave_gemm_mt(int K, FA loadA, FB loadB, FS store) {
  const int lane = threadIdx.x & 31;
  const int half = lane >> 4;
  const int l16  = lane & 15;
  v8f acc[MT];
#pragma unroll
  for (int t = 0; t < MT; ++t)
#pragma unroll
    for (int i = 0; i < 8; ++i) acc[t][i] = 0.0f;

  for (int k0 = 0; k0 < K; k0 += 4) {
    const int kk = k0 + 2 * half;
    v2f b = loadB(l16, kk);
#pragma unroll
    for (int t = 0; t < MT; ++t) {
      v2f a = loadA(t, l16, kk);
      acc[t] = __builtin_amdgcn_wmma_f32_16x16x4_f32(false, a, false, b,
                                                     (short)0, acc[t],
                                                     false, false);
    }
  }
#pragma unroll
  for (int t = 0; t < MT; ++t)
#pragma unroll
    for (int r = 0; r < 8; ++r) store(t, r + 8 * half, l16, acc[t][r]);
}

// ---------------------------------------------------------------------------
// 1) Antialiased bilinear resize (B,S,64,64) -> (B,S,16,24), jax semantics:
//    sample = (out+0.5)/scale - 0.5, triangle kernel stretched by 1/scale,
//    per-axis weight normalization (== joint normalization for separable tri).
// ---------------------------------------------------------------------------
__global__ void resize_kernel(const float* __restrict__ masks, float* __restrict__ m) {
  int idx = blockIdx.x * blockDim.x + threadIdx.x;
  if (idx >= B_ * S_ * HW_) return;
  int w  = idx % W_;
  int h  = (idx / W_) % H_;
  int bs = idx / HW_;
  const float* src = masks + (size_t)bs * HM_ * WM_;

  const float sh = (float)H_ / (float)HM_;   // 0.25
  const float sw = (float)W_ / (float)WM_;   // 0.375
  float ph = (h + 0.5f) / sh - 0.5f;
  float pw = (w + 0.5f) / sw - 0.5f;
  int h0 = max(0,       (int)ceilf (ph - 1.0f / sh));
  int h1 = min(HM_ - 1, (int)floorf(ph + 1.0f / sh));
  int w0 = max(0,       (int)ceilf (pw - 1.0f / sw));
  int w1 = min(WM_ - 1, (int)floorf(pw + 1.0f / sw));

  float acc = 0.0f, wsum = 0.0f;
  for (int hh = h0; hh <= h1; ++hh) {
    float wh = 1.0f - fabsf((hh - ph) * sh);
    if (wh <= 0.0f) continue;
    for (int ww = w0; ww <= w1; ++ww) {
      float wv = 1.0f - fabsf((ww - pw) * sw);
      if (wv <= 0.0f) continue;
      float wt = wh * wv;
      acc  += wt * src[hh * WM_ + ww];
      wsum += wt;
    }
  }
  m[idx] = acc / wsum;
}

// ---------------------------------------------------------------------------
// 2) conv1x1 as GEMM: out[b,o,hw] = sum_c w[o,c] * x[b,c,hw] + bias[o]
//    N = B*HW = 768 columns, K = 512, M = 512 (ctx) or 768 (rel).
//    One wave computes a 64x16 output block (MT=4).
// ---------------------------------------------------------------------------
__global__ void conv1x1_kernel(const float* __restrict__ x,
                               const float* __restrict__ w,
                               const float* __restrict__ bias,
                               float* __restrict__ out, int M) {
  const int ntiles = (B_ * HW_) / 16;              // 48
  const int total  = (M / 64) * ntiles;
  int wave = (blockIdx.x * blockDim.x + threadIdx.x) >> 5;
  if (wave >= total) return;                       // wave-uniform
  const int mbase = (wave / ntiles) * 64;
  const int nbase = (wave % ntiles) * 16;

  wave_gemm_mt<4>(CIN_,
    [=](int t, int r, int k) -> v2f {
      const float* p = w + (size_t)(mbase + t * 16 + r) * CIN_ + k;
      v2f a; a.x = p[0]; a.y = p[1]; return a;
    },
    [=](int c, int k) -> v2f {
      int n = nbase + c, b = n / HW_, hw = n % HW_;
      const float* p = x + ((size_t)b * CIN_ + k) * HW_ + hw;
      v2f v; v.x = p[0]; v.y = p[HW_]; return v;
    },
    [=](int t, int r, int c, float v) {
      int n = nbase + c, b = n / HW_, hw = n % HW_;
      int o = mbase + t * 16 + r;
      out[((size_t)b * M + o) * HW_ + hw] = v + bias[o];
    });
}

// ---------------------------------------------------------------------------
// 3) obj_feat[b,s,o,hw] = sum_k w_obj[o,k] * comb_obj[k] + b_obj[o]
//    comb_obj materialized in the B-fragment loader (amortized over 4 WMMAs):
//      k<512 : input*m | k<768 : ctx*m | k<1024 : ctx*m^2 | else m*w_mask+b_mask
//    (region boundaries even, per-lane k is even -> pair never straddles)
// ---------------------------------------------------------------------------
__global__ void obj_fuse_kernel(const float* __restrict__ input,
                                const float* __restrict__ ctx,
                                const float* __restrict__ m,
                                const float* __restrict__ w,
                                const float* __restrict__ bias,
                                const float* __restrict__ w_mask,
                                const float* __restrict__ b_mask,
                                float* __restrict__ objf) {
  const int tiles_per = (D_ / 64) * (HW_ / 16);    // 4*24 = 96
  int wave = (blockIdx.x * blockDim.x + threadIdx.x) >> 5;
  if (wave >= B_ * S_ * tiles_per) return;
  int bs = wave / tiles_per, t0 = wave % tiles_per;
  const int mbase = (t0 / (HW_ / 16)) * 64;
  const int nbase = (t0 % (HW_ / 16)) * 16;
  const int b = bs / S_;
  const float* mrow = m + (size_t)bs * HW_;

  wave_gemm_mt<4>(COBJ_,
    [=](int t, int r, int k) -> v2f {
      const float* p = w + (size_t)(mbase + t * 16 + r) * COBJ_ + k;
      v2f a; a.x = p[0]; a.y = p[1]; return a;
    },
    [=](int c, int k) -> v2f {
      int hw = nbase + c;
      float mm = mrow[hw];
      v2f v;
      if (k < 512) {
        const float* p = input + ((size_t)b * CIN_ + k) * HW_ + hw;
        v.x = p[0] * mm; v.y = p[HW_] * mm;
      } else if (k < 768) {
        const float* p = ctx + ((size_t)b * CIN_ + (k - 512)) * HW_ + hw;
        v.x = p[0] * mm; v.y = p[HW_] * mm;
      } else if (k < 1024) {
        const float* p = ctx + ((size_t)b * CIN_ + (k - 512)) * HW_ + hw;
        float m2 = mm * mm;
        v.x = p[0] * m2; v.y = p[HW_] * m2;
      } else {
        int mf = k - 1024;
        v.x = mm * w_mask[mf]     + b_mask[mf];
        v.y = mm * w_mask[mf + 1] + b_mask[mf + 1];
      }
      return v;
    },
    [=](int t, int r, int c, float v) {
      int hw = nbase + c, o = mbase + t * 16 + r;
      objf[((size_t)bs * D_ + o) * HW_ + hw] = v + bias[o];
    });
}

// ---------------------------------------------------------------------------
// 4) rel_feat (stored post-ReLU; it is only consumed through ReLU).
//    Dominant GEMM (~64 GFLOP). comb_rel regions:
//    objf[sub] | objf[obj] | rel*pm | rel*pm*ms | rel*pm*mo | mf[sub] | mf[obj]
// ---------------------------------------------------------------------------
__global__ void rel_fuse_kernel(const float* __restrict__ objf,
                                const float* __restrict__ rel,
                                const float* __restrict__ m,
                                const float* __restrict__ w,
                                const float* __restrict__ bias,
                                const float* __restrict__ w_mask,
                                const float* __restrict__ b_mask,
                                float* __restrict__ relf) {
  const int tiles_per = (D_ / 64) * (HW_ / 16);    // 96
  int wave = (blockIdx.x * blockDim.x + threadIdx.x) >> 5;
  if (wave >= B_ * S_ * S_ * tiles_per) return;
  int bp = wave / tiles_per, t0 = wave % tiles_per;
  const int mbase = (t0 / (HW_ / 16)) * 64;
  const int nbase = (t0 % (HW_ / 16)) * 16;
  const int b = bp / (S_ * S_), p = bp % (S_ * S_);
  const int sub = p / S_, obj = p % S_;
  const float* msub = m + (size_t)(b * S_ + sub) * HW_;
  const float* mobj = m + (size_t)(b * S_ + obj) * HW_;
  const float* fsub = objf + (size_t)(b * S_ + sub) * D_ * HW_;
  const float* fobj = objf + (size_t)(b * S_ + obj) * D_ * HW_;

  wave_gemm_mt<4>(CRIN_,
    [=](int t, int r, int k) -> v2f {
      const float* q = w + (size_t)(mbase + t * 16 + r) * CRIN_ + k;
      v2f a; a.x = q[0]; a.y = q[1]; return a;
    },
    [=](int c, int k) -> v2f {
      int hw = nbase + c;
      float ms = msub[hw], mo = mobj[hw], pm = ms + mo;
      v2f v;
      if (k < 256) {
        const float* q = fsub + (size_t)k * HW_ + hw;
        v.x = q[0]; v.y = q[HW_];
      } else if (k < 512) {
        const float* q = fobj + (size_t)(k - 256) * HW_ + hw;
        v.x = q[0]; v.y = q[HW_];
      } else if (k < 768) {
        const float* q = rel + ((size_t)b * CREL_ + (k - 512)) * HW_ + hw;
        v.x = q[0] * pm; v.y = q[HW_] * pm;
      } else if (k < 1024) {
        const float* q = rel + ((size_t)b * CREL_ + (k - 512)) * HW_ + hw;
        float f = pm * ms;
        v.x = q[0] * f; v.y = q[HW_] * f;
      } else if (k < 1280) {
        const float* q = rel + ((size_t)b * CREL_ + (k - 512)) * HW_ + hw;
        float f = pm * mo;
        v.x = q[0] * f; v.y = q[HW_] * f;
      } else if (k < 1312) {
        int mf = k - 1280;
        v.x = ms * w_mask[mf]     + b_mask[mf];
        v.y = ms * w_mask[mf + 1] + b_mask[mf + 1];
      } else {
        int mf = k - 1312;
        v.x = mo * w_mask[mf]     + b_mask[mf];
        v.y = mo * w_mask[mf + 1] + b_mask[mf + 1];
      }
      return v;
    },
    [=](int t, int r, int c, float v) {
      int hw = nbase + c, o = mbase + t * 16 + r;
      relf[((size_t)bp * D_ + o) * HW_ + hw] = fmaxf(v + bias[o], 0.0f);
    });
}

// ---------------------------------------------------------------------------
// 5) FC: fc[row,o] = sum_k w_fc[o,k] * relu(feat[row,k]) + b_fc[o]
//    A = w_fc (M=256 x K=98304), B = relu(feat)^T; ReLU is idempotent so it
//    is applied on both the raw obj path and the already-ReLU'd rel path.
// ---------------------------------------------------------------------------
__global__ void fc_kernel(const float* __restrict__ feat,
                          const float* __restrict__ w_fc,
                          const float* __restrict__ b_fc,
                          float* __restrict__ out, int nrows) {
  const int ntiles = (nrows + 15) / 16;
  int wave = (blockIdx.x * blockDim.x + threadIdx.x) >> 5;
  if (wave >= (D_ / 64) * ntiles) return;
  const int mbase = (wave / ntiles) * 64;
  const int nbase = (wave % ntiles) * 16;

  wave_gemm_mt<4>(FCK_,
    [=](int t, int r, int k) -> v2f {
      const float* p = w_fc + (size_t)(mbase + t * 16 + r) * FCK_ + k;
      v2f a; a.x = p[0]; a.y = p[1]; return a;
    },
    [=](int c, int k) -> v2f {
      int row = nbase + c;
      v2f v;
      if (row < nrows) {
        const float* p = feat + (size_t)row * FCK_ + k;
        v.x = fmaxf(p[0], 0.0f); v.y = fmaxf(p[1], 0.0f);
      } else {
        v.x = 0.0f; v.y = 0.0f;
      }
      return v;
    },
    [=](int t, int r, int c, float v) {
      int row = nbase + c;
      if (row < nrows) {
        int o = mbase + t * 16 + r;
        out[(size_t)row * D_ + o] = v + b_fc[o];
      }
    });
}

// ---------------------------------------------------------------------------
// 6) L2 normalize each 256-row of fcbuf into d_out (obj rows then rel rows).
// ---------------------------------------------------------------------------
__global__ void l2norm_kernel(const float* __restrict__ fcbuf, float* __restrict__ out) {
  __shared__ float red[256];
  int row = blockIdx.x, t = threadIdx.x;
  float v = fcbuf[(size_t)row * D_ + t];
  red[t] = v * v;
  __syncthreads();
  for (int s = 128; s > 0; s >>= 1) {
    if (t < s) red[t] += red[t + s];
    __syncthreads();
  }
  out[(size_t)row * D_ + t] = v / sqrtf(red[0]);
}

// ---------------------------------------------------------------------------
// Workspace layout (floats):
//   m       8448
//   ctx     393216
//   rel     589824
//   objf    2162688
//   relf    23789568   (95 MB intermediate; fits in the 192 MB L2)
//   fcbuf   67584
//   total  27,011,328 floats = ~103 MB
// ---------------------------------------------------------------------------
extern "C" void kernel_launch(void* const* d_in, const int* in_sizes, int n_in,
                              void* d_out, int out_size, void* d_ws, size_t ws_size,
                              hipStream_t stream) {
  const float* input      = (const float*)d_in[0];
  const float* masks      = (const float*)d_in[1];
  const float* w_ctx      = (const float*)d_in[2];
  const float* b_ctx      = (const float*)d_in[3];
  const float* w_rel      = (const float*)d_in[4];
  const float* b_rel      = (const float*)d_in[5];
  const float* w_mask     = (const float*)d_in[6];
  const float* b_mask     = (const float*)d_in[7];
  const float* w_obj_fuse = (const float*)d_in[8];
  const float* b_obj_fuse = (const float*)d_in[9];
  const float* w_rel_fuse = (const float*)d_in[10];
  const float* b_rel_fuse = (const float*)d_in[11];
  const float* w_fc       = (const float*)d_in[12];
  const float* b_fc       = (const float*)d_in[13];

  float* ws    = (float*)d_ws;
  float* m     = ws;
  float* ctx   = m + 8448;
  float* rel   = ctx + 393216;
  float* objf  = rel + 589824;
  float* relf  = objf + 2162688;
  float* fcbuf = relf + 23789568;

  // 1) mask resize
  resize_kernel<<<(B_ * S_ * HW_ + 255) / 256, 256, 0, stream>>>(masks, m);

  // 2) conv1x1 context (8x48 = 384 wave-blocks) and relation (12x48 = 576)
  conv1x1_kernel<<<48, 256, 0, stream>>>(input, w_ctx, b_ctx, ctx, CIN_);
  conv1x1_kernel<<<72, 256, 0, stream>>>(input, w_rel, b_rel, rel, CREL_);

  // 3) obj_fuse: 22 * 96 = 2112 wave-blocks
  obj_fuse_kernel<<<264, 256, 0, stream>>>(input, ctx, m, w_obj_fuse, b_obj_fuse,
                                           w_mask, b_mask, objf);

  // 4) rel_fuse (dominant, ~64 GFLOP): 242 * 96 = 23232 wave-blocks
  rel_fuse_kernel<<<2904, 256, 0, stream>>>(objf, rel, m, w_rel_fuse, b_rel_fuse,
                                            w_mask, b_mask, relf);

  // 5) FC: obj rows (4x2 = 8 waves), rel rows (4x16 = 64 waves)
  fc_kernel<<<1, 256, 0, stream>>>(objf, w_fc, b_fc, fcbuf, B_ * S_);
  fc_kernel<<<8, 256, 0, stream>>>(relf, w_fc, b_fc, fcbuf + (size_t)B_ * S_ * D_,
                                   B_ * S_ * S_);

  // 6) L2 norm -> d_out (obj_out flat then rel_out flat)
  l2norm_kernel<<<264, 256, 0, stream>>>(fcbuf, (float*)d_out);
}